// PDFSampler_76433238000154
// MI455X (gfx1250) — compile-verified
//
#include <hip/hip_runtime.h>
#include <stdint.h>

// PDF inverse-transform resampling (NeRF sample_pdf), R rays:
//   weights:       [R, 64]  f32
//   existing_bins: [R, 65]  f32 (sorted per ray)
//   out:           [R, 130] f32 (merge of 65 existing edges + 65 samples, sorted)
//
// Memory-bound: ~34 MB total traffic -> ~1.5 us at 23.3 TB/s. Algorithm is
// O(N) per ray (two-pointer CDF scan + sorted merge) instead of the reference's
// O(N^2) broadcast mask, so VALU cost is negligible; the win is data movement.
// existing_bins is staged via CDNA5 async global->LDS DMA (ASYNCcnt) into a
// column-per-thread LDS layout (bank-conflict-free under divergent indexing),
// overlapped with per-thread register-resident CDF construction.

#define BLOCK    64      // threads per block = rays per block (2 waves, wave32)
#define NW       64      // weights per ray
#define NE       65      // bin edges per ray (= cdf length)
#define NS       65      // probe samples per ray
#define NOUT     130     // merged output per ray
#define HPAD     0.01f
#define EPSP     1e-5f

__global__ __launch_bounds__(BLOCK)
void pdf_sampler_kernel(const float* __restrict__ weights,
                        const float* __restrict__ ebins,
                        float* __restrict__ out,
                        int R)
{
    // Column-per-thread layout: element i of thread t at [i*BLOCK + t].
    // Stride 64 words == 64 banks -> bank depends only on t: conflict-free
    // even when threads diverge in i (interval search / merge).
    __shared__ float sCdf[NE * BLOCK];
    __shared__ float sEb [NE * BLOCK];

    const int tid     = threadIdx.x;
    const int rayBase = blockIdx.x * BLOCK;
    const int ray     = rayBase + tid;

    // ---- 1. Async-DMA existing_bins for this block into LDS (transposed) ----
    {
        const int nRays = (R - rayBase < BLOCK) ? (R - rayBase) : BLOCK;
        const int N     = nRays * NE;                       // floats in region
        const uint32_t ldsBase = (uint32_t)(uintptr_t)&sEb[0];  // low 32 bits = LDS offset
        const unsigned long long gbase =
            (unsigned long long)(const void*)(ebins + (size_t)rayBase * NE);
        for (int idx = tid; idx < N; idx += BLOCK) {
            int r = idx / NE;                // ray within block
            int i = idx - r * NE;            // edge index
            uint32_t ldsAddr = ldsBase + (uint32_t)((i * BLOCK + r) * 4);
            uint32_t goff    = (uint32_t)(idx * 4);         // coalesced across lanes
            // GVS mode: mem = SGPR base + 32-bit VGPR offset. Tracked by ASYNCcnt.
            asm volatile("global_load_async_to_lds_b32 %0, %1, %2"
                         :: "v"(ldsAddr), "v"(goff), "s"(gbase)
                         : "memory");
        }
    }

    // ---- 2. Overlap: per-thread weights -> padded pdf -> clamped cumsum ----
    if (ray < R) {
        const float4* wv = (const float4*)(weights + (size_t)ray * NW);
        float4 w[NW / 4];
        #pragma unroll
        for (int c = 0; c < NW / 4; ++c) w[c] = wv[c];      // 16x global_load_b128

        float s = 0.f;
        #pragma unroll
        for (int c = 0; c < NW / 4; ++c)
            s += (w[c].x + HPAD) + (w[c].y + HPAD) + (w[c].z + HPAD) + (w[c].w + HPAD);

        float pad = fmaxf(EPSP - s, 0.f);
        float inv = 1.f / (s + pad);
        float add = pad * (1.f / NW);

        float run = 0.f;
        sCdf[tid] = 0.f;                                    // cdf[0] = 0
        #pragma unroll
        for (int c = 0; c < NW / 4; ++c) {
            run += (w[c].x + HPAD + add) * inv;  sCdf[(4*c + 1) * BLOCK + tid] = fminf(run, 1.f);
            run += (w[c].y + HPAD + add) * inv;  sCdf[(4*c + 2) * BLOCK + tid] = fminf(run, 1.f);
            run += (w[c].z + HPAD + add) * inv;  sCdf[(4*c + 3) * BLOCK + tid] = fminf(run, 1.f);
            run += (w[c].w + HPAD + add) * inv;  sCdf[(4*c + 4) * BLOCK + tid] = fminf(run, 1.f);
        }
    }

    // Wave waits for its own DMA, barrier publishes every wave's portion.
    asm volatile("s_wait_asynccnt 0" ::: "memory");
    __syncthreads();

    if (ray >= R) return;

    // ---- 3. Fused inverse-CDF sampling + sorted merge (130 outputs) ----
    float* __restrict__ outp = out + (size_t)ray * NOUT;

    int below = 0;                 // max i with cdf[i] <= u_j  (cdf monotone, u increasing)
    int j = 0;                     // current sample index, value held in `samp`
    int a = 0;                     // current existing-edge index, value in `eb`
    // j=0: u=0, below=0 (cdf[1] > 0 since pdf > 0), t=0 -> samp = ebins[0]
    float samp = sEb[tid];
    float eb   = sEb[tid];

    for (int k = 0; k < NOUT; ++k) {
        bool takeE = (j >= NS) || ((a < NE) && (eb <= samp));
        outp[k] = takeE ? eb : samp;
        if (takeE) {
            ++a;
            eb = (a < NE) ? sEb[a * BLOCK + tid] : __builtin_inff();
        } else {
            ++j;
            if (j < NS) {
                float u = (float)j * 0.015625f;             // j/64, exact
                while (below < NW && sCdf[(below + 1) * BLOCK + tid] <= u) ++below;
                int   ab = (below < NW) ? below + 1 : NW;   // clamp to index 64
                float c0 = sCdf[below * BLOCK + tid];
                float c1 = sCdf[ab    * BLOCK + tid];
                float b0 = sEb [below * BLOCK + tid];
                float b1 = sEb [ab    * BLOCK + tid];
                float d   = c1 - c0;                        // >= 0
                float num = u - c0;                         // >= 0 by def of below
                // matches clip(nan_to_num(num/d), 0, 1): d==0 -> nan->0 / inf->1
                float t = (d > 0.f) ? fminf(fmaxf(num / d, 0.f), 1.f)
                                    : ((num > 0.f) ? 1.f : 0.f);
                samp = b0 + t * (b1 - b0);
            }
        }
    }
}

extern "C" void kernel_launch(void* const* d_in, const int* in_sizes, int n_in,
                              void* d_out, int out_size, void* d_ws, size_t ws_size,
                              hipStream_t stream) {
    const float* weights = (const float*)d_in[0];   // [R, 64]
    const float* ebins   = (const float*)d_in[1];   // [R, 65]
    float*       out     = (float*)d_out;           // [R, 130]
    const int R = in_sizes[0] / NW;
    const int grid = (R + BLOCK - 1) / BLOCK;
    pdf_sampler_kernel<<<grid, BLOCK, 0, stream>>>(weights, ebins, out, R);
}